// Net_77644418777677
// MI455X (gfx1250) — compile-verified
//
#include <hip/hip_runtime.h>

typedef float v2f __attribute__((ext_vector_type(2)));
typedef float v8f __attribute__((ext_vector_type(8)));

#define M_ROWS 262144
#define NCOLS  128
#define H1DIM  16
#define NCDIM  4

#define WAVES_PER_BLOCK 2
#define BLOCK_THREADS   (WAVES_PER_BLOCK * 32)
#define W_STRIDE 129   // padded LDS row stride for w tiles: bank = (lane + k) % 64, conflict-free
#define H_STRIDE 20    // padded LDS row stride for h tiles: lane halves hit disjoint banks

__global__ __launch_bounds__(BLOCK_THREADS)
void fused_ddk_mlp(const float* __restrict__ x,
                   const float* __restrict__ alpha,
                   const float* __restrict__ beta,
                   const float* __restrict__ W1,
                   const float* __restrict__ b1,
                   const float* __restrict__ W2,
                   const float* __restrict__ b2,
                   const int*   __restrict__ target,
                   float* __restrict__ out,      // M x 4
                   float* __restrict__ out_tgt)  // M
{
    extern __shared__ float smem[];
    float* wbuf = smem;                                        // [WAVES][32][W_STRIDE]
    float* Wt   = smem + WAVES_PER_BLOCK * 32 * W_STRIDE;      // [128][16]  W1 transposed
    float* hbuf = Wt + NCOLS * H1DIM;                          // [WAVES][32][H_STRIDE]

    const int tid  = threadIdx.x;
    const int wave = tid >> 5;
    const int lane = tid & 31;
    const int row  = blockIdx.x * BLOCK_THREADS + tid;

    // ---- stage W1 transposed into LDS: Wt[k][n] = W1[n][k] (coalesced global reads) ----
    for (int f = tid; f < H1DIM * NCOLS; f += BLOCK_THREADS) {
        int n = f >> 7;       // / 128
        int k = f & 127;
        Wt[k * H1DIM + n] = W1[f];
    }

    const float a = alpha[0];
    const float b = beta[0];

    // ---- phase 1: recurrence, one row per lane, x streamed as float4 ----
    const float4* xr = (const float4*)(x + (size_t)row * NCOLS);
    float* wrow = wbuf + (size_t)tid * W_STRIDE;

    float w = 1.0f;
    wrow[0] = w;
    #pragma unroll 4
    for (int k4 = 0; k4 < 31; ++k4) {
        float4 xv = xr[k4];
        w = w + a * __builtin_amdgcn_rcpf(w) - b * xv.x; wrow[4 * k4 + 1] = w;
        w = w + a * __builtin_amdgcn_rcpf(w) - b * xv.y; wrow[4 * k4 + 2] = w;
        w = w + a * __builtin_amdgcn_rcpf(w) - b * xv.z; wrow[4 * k4 + 3] = w;
        w = w + a * __builtin_amdgcn_rcpf(w) - b * xv.w; wrow[4 * k4 + 4] = w;
    }
    {   // tail: k = 124..126 produce w[125..127]
        float4 xv = xr[31];
        w = w + a * __builtin_amdgcn_rcpf(w) - b * xv.x; wrow[125] = w;
        w = w + a * __builtin_amdgcn_rcpf(w) - b * xv.y; wrow[126] = w;
        w = w + a * __builtin_amdgcn_rcpf(w) - b * xv.z; wrow[127] = w;
    }
    __syncthreads();

    // ---- phase 2: h = sigmoid(w @ W1^T + b1) via V_WMMA_F32_16X16X4_F32 ----
    // A 16x4 f32 layout: lanes 0-15 hold M=0..15 with VGPR0=K0,VGPR1=K1; lanes 16-31: K2,K3
    // B 4x16 f32 layout: N across lanes; VGPR0 = K row 0 (lower half)/row 2 (upper); VGPR1 = 1/3
    const int nn   = lane & 15;            // N index (neuron) and A row within tile
    const int koff = (lane >> 4) * 2;      // K sub-offset by lane half
    const float* wtile = wbuf + (size_t)wave * 32 * W_STRIDE;
    const float b1n = b1[nn];

    #pragma unroll
    for (int tile = 0; tile < 2; ++tile) {
        const float* arow = wtile + (size_t)(tile * 16 + nn) * W_STRIDE;
        v8f c = {0.f, 0.f, 0.f, 0.f, 0.f, 0.f, 0.f, 0.f};
        #pragma unroll
        for (int kb = 0; kb < NCOLS; kb += 4) {
            v2f av, bv;
            av.x = arow[kb + koff];
            av.y = arow[kb + koff + 1];
            bv.x = Wt[(kb + koff)     * H1DIM + nn];
            bv.y = Wt[(kb + koff + 1) * H1DIM + nn];
            c = __builtin_amdgcn_wmma_f32_16x16x4_f32(
                    false, av, false, bv, (short)0, c, false, false);
        }
        // C layout: VGPR v -> row m = v + 8*(lane>=16), col = lane&15
        #pragma unroll
        for (int v = 0; v < 8; ++v) {
            int   m   = v + 8 * (lane >> 4);
            float pre = c[v] + b1n;
            float hs  = __builtin_amdgcn_rcpf(
                            1.0f + __builtin_amdgcn_exp2f(-1.4426950408889634f * pre));
            hbuf[((size_t)wave * 32 + tile * 16 + m) * H_STRIDE + nn] = hs;
        }
    }
    __syncthreads();

    // ---- phase 3: out = h @ W2^T + b2 (tiny K=16 dot per row, one row per lane) ----
    const float* hrow = hbuf + (size_t)tid * H_STRIDE;
    float acc0 = b2[0], acc1 = b2[1], acc2 = b2[2], acc3 = b2[3];
    #pragma unroll
    for (int j = 0; j < H1DIM; ++j) {
        float hv = hrow[j];
        acc0 += hv * W2[0 * H1DIM + j];
        acc1 += hv * W2[1 * H1DIM + j];
        acc2 += hv * W2[2 * H1DIM + j];
        acc3 += hv * W2[3 * H1DIM + j];
    }
    float4 o = make_float4(acc0, acc1, acc2, acc3);
    ((float4*)out)[row] = o;

    // target passthrough (second tuple element), stored as float
    out_tgt[row] = (float)target[row];
}

extern "C" void kernel_launch(void* const* d_in, const int* in_sizes, int n_in,
                              void* d_out, int out_size, void* d_ws, size_t ws_size,
                              hipStream_t stream) {
    const float* x      = (const float*)d_in[0];
    const float* alpha  = (const float*)d_in[1];
    const float* beta   = (const float*)d_in[2];
    const float* W1     = (const float*)d_in[3];
    const float* b1     = (const float*)d_in[4];
    const float* W2     = (const float*)d_in[5];
    const float* b2     = (const float*)d_in[6];
    const int*   target = (const int*)d_in[7];

    float* out     = (float*)d_out;
    float* out_tgt = out + (size_t)M_ROWS * NCDIM;

    const int blocks = M_ROWS / BLOCK_THREADS;   // exact: 262144 / 64 = 4096
    const size_t lds_bytes =
        (size_t)(WAVES_PER_BLOCK * 32 * W_STRIDE + NCOLS * H1DIM +
                 WAVES_PER_BLOCK * 32 * H_STRIDE) * sizeof(float);   // ~46 KB

    fused_ddk_mlp<<<blocks, BLOCK_THREADS, lds_bytes, stream>>>(
        x, alpha, beta, W1, b1, W2, b2, target, out, out_tgt);
}